// SubtrajectoryBalanceLoss_87024627352242
// MI455X (gfx1250) — compile-verified
//
#include <hip/hip_runtime.h>

// SubTB loss, algebraically reduced from O(B*S^2) pairs to O(B*S):
//   a[k] = clip(log(max(ps*v,eps)),-10,10) + cum_lp_excl[k]
//   pair term (m,n) = (a[m]-a[n])^2 ; valid pairs = all pairs inside each
//   mask-run segment (run of ones + trailing boundary index, capped at S-1).
//   Per segment: sum = L*Sum(a^2) - (Sum a)^2 ; pairs = L*(L-1)/2.
// Segment id: sid[k] = k - (#ones before k)  (monotone; masked-out singletons
// contribute exactly zero, so no special-casing needed).
//
// wave32-optimized: scans use 5-step __shfl_up intra-wave scans + one LDS
// exchange of the 32 wave sums (6 block barriers total instead of ~60);
// final 1024-wide reductions run through chained V_WMMA_F32_16X16X4_F32.

#define SEQ 1024
#define BLK 1024
#define NW  (BLK / 32)

typedef float v2f __attribute__((ext_vector_type(2)));
typedef float v8f __attribute__((ext_vector_type(8)));

// Sum buf[0..1023] cooperatively within one wave using V_WMMA_F32_16X16X4_F32.
// A = all-ones, so D accumulates column sums of B; a grand total is
// layout-agnostic: Sum(all D elements) = 16 * Sum(all B elements fed in).
// Every wave of the block calls this (EXEC all ones, as WMMA requires);
// all waves compute the identical value.
__device__ __forceinline__ float wmma_reduce_1024(const float* __restrict__ buf,
                                                  int lane) {
    v2f a; a.x = 1.0f; a.y = 1.0f;          // ones matrix (any layout)
    v8f c = {};
    #pragma unroll
    for (int it = 0; it < 16; ++it) {
        v2f bm;
        bm.x = buf[it * 64 + lane];
        bm.y = buf[it * 64 + 32 + lane];
        c = __builtin_amdgcn_wmma_f32_16x16x4_f32(
                /*neg_a=*/false, a, /*neg_b=*/false, bm,
                /*c_mod=*/(short)0, c, /*reuse_a=*/false, /*reuse_b=*/false);
    }
    float s = c[0] + c[1] + c[2] + c[3] + c[4] + c[5] + c[6] + c[7];
    #pragma unroll
    for (int m = 16; m >= 1; m >>= 1) s += __shfl_xor(s, m, 32);  // wave32
    return s * (1.0f / 16.0f);   // 16 identical rows in D
}

// Inclusive intra-wave scan of two values (5 shuffle steps, no LDS).
__device__ __forceinline__ void wave_scan2(float& x0, float& x1, int lane) {
    #pragma unroll
    for (int d = 1; d < 32; d <<= 1) {
        float t0 = __shfl_up(x0, d, 32);
        float t1 = __shfl_up(x1, d, 32);
        if (lane >= d) { x0 += t0; x1 += t1; }
    }
}

__global__ __launch_bounds__(BLK)
void subtb_batch_kernel(const float* __restrict__ lp,
                        const float* __restrict__ vals,
                        const float* __restrict__ ps,
                        const int*   __restrict__ mask,
                        float* __restrict__ ws) {
    __shared__ float wS0[NW];
    __shared__ float wS1[NW];
    __shared__ float wR[NW];
    __shared__ float CA[BLK];
    __shared__ float CA2[BLK];
    __shared__ int   sidArr[BLK];
    __shared__ int   segStart[BLK];
    __shared__ float contrib[BLK];
    __shared__ float pairsArr[BLK];

    const int b    = blockIdx.x;
    const int k    = threadIdx.x;
    const int lane = k & 31;
    const int w    = k >> 5;
    const int base = b * SEQ;

    const float myLp = lp[base + k];
    const float v    = vals[base + k];
    const float p    = ps[base + k];
    const int   mk   = (mask[base + k] != 0) ? 1 : 0;

    const float flow = fmaxf(p * v, 1e-8f);
    const float lf   = fminf(fmaxf(logf(flow), -10.0f), 10.0f);

    // ---- Block-wide joint inclusive scan of (policy_logprob, mask) ----
    float x0 = myLp;
    float x1 = (float)mk;                 // mask sums <= 1024, exact in f32
    wave_scan2(x0, x1, lane);
    if (lane == 31) { wS0[w] = x0; wS1[w] = x1; }
    __syncthreads();
    {
        float c0 = 0.0f, c1 = 0.0f;
        for (int i = 0; i < w; ++i) { c0 += wS0[i]; c1 += wS1[i]; }
        x0 += c0; x1 += c1;               // inclusive block scan
    }
    const float cumLpExcl = x0 - myLp;
    const int   cumMExcl  = (int)(x1 + 0.5f) - mk;
    const int   sid       = k - cumMExcl;          // #zeros strictly before k

    const float aval = lf + cumLpExcl;

    // ---- Row mean of a (pairwise differences are shift-invariant) ----
    {
        float s = aval;
        #pragma unroll
        for (int m = 16; m >= 1; m >>= 1) s += __shfl_xor(s, m, 32);
        if (lane == 0) wR[w] = s;
    }
    __syncthreads();                      // also fences wS0/wS1 reads above
    float mean;
    {
        float tot = 0.0f;
        #pragma unroll
        for (int i = 0; i < NW; ++i) tot += wR[i];   // same order in all threads
        mean = tot * (1.0f / SEQ);
    }
    const float ac = aval - mean;

    // ---- Block-wide joint inclusive scan of (a', a'^2) ----
    float y0 = ac;
    float y1 = ac * ac;
    wave_scan2(y0, y1, lane);
    if (lane == 31) { wS0[w] = y0; wS1[w] = y1; }
    __syncthreads();
    {
        float c0 = 0.0f, c1 = 0.0f;
        for (int i = 0; i < w; ++i) { c0 += wS0[i]; c1 += wS1[i]; }
        y0 += c0; y1 += c1;
    }
    CA[k]     = y0;
    CA2[k]    = y1;
    sidArr[k] = sid;
    __syncthreads();

    // ---- Segment boundaries and per-segment closed-form contribution ----
    const bool isStart = (k == 0)       || (sidArr[k - 1] != sid);
    const bool isEnd   = (k == BLK - 1) || (sidArr[k + 1] != sid);
    if (isStart) segStart[sid] = k;       // one writer per sid
    __syncthreads();

    float ctb = 0.0f, prs = 0.0f;
    if (isEnd) {
        const int   s0    = segStart[sid];
        const float Lf    = (float)(k - s0 + 1);
        const float sumA  = CA[k]  - (s0 > 0 ? CA[s0 - 1]  : 0.0f);
        const float sumA2 = CA2[k] - (s0 > 0 ? CA2[s0 - 1] : 0.0f);
        ctb = fmaxf(Lf * sumA2 - sumA * sumA, 0.0f);
        prs = 0.5f * Lf * (Lf - 1.0f);    // pair counts < 2^24: exact in f32
    }
    contrib[k]  = ctb;
    pairsArr[k] = prs;
    __syncthreads();

    // ---- WMMA block reductions (every wave runs them: EXEC all ones) ----
    const float total = wmma_reduce_1024(contrib,  lane);
    const float num   = wmma_reduce_1024(pairsArr, lane);
    if (k == 0) ws[b] = total / fmaxf(num, 1.0f);
}

__global__ void subtb_final_kernel(const float* __restrict__ ws,
                                   float* __restrict__ out, int B) {
    const int k = threadIdx.x;
    float x = 0.0f;
    for (int i = k; i < B; i += 32) x += ws[i];
    #pragma unroll
    for (int m = 16; m >= 1; m >>= 1) x += __shfl_xor(x, m, 32);
    if (k == 0) out[0] = x / (float)B;
}

extern "C" void kernel_launch(void* const* d_in, const int* in_sizes, int n_in,
                              void* d_out, int out_size, void* d_ws, size_t ws_size,
                              hipStream_t stream) {
    const float* lp   = (const float*)d_in[0];   // policy_logprobs [B,S] f32
    const float* vals = (const float*)d_in[1];   // values          [B,S] f32
    const float* ps   = (const float*)d_in[2];   // prefix_scores   [B,S] f32
    const int*   mk   = (const int*)d_in[3];     // attention_mask  [B,S] bool->int
    float* out = (float*)d_out;                  // scalar loss f32
    float* ws  = (float*)d_ws;                   // [B] per-batch normalized loss

    const int B = in_sizes[0] / SEQ;
    subtb_batch_kernel<<<B, BLK, 0, stream>>>(lp, vals, ps, mk, ws);
    subtb_final_kernel<<<1, 32, 0, stream>>>(ws, out, B);
}